// Model_84817014162198
// MI455X (gfx1250) — compile-verified
//
#include <hip/hip_runtime.h>

// ---------------------------------------------------------------------------
// Heterogeneous 2-layer GraphSAGE for MI455X (gfx1250, wave32).
//   encode   : per-type 32->32 feature GEMV + ReLU, concat with embedding
//   scatter  : per-edge gather + f32 atomic segment-sum (L2-resident tables)
//   finalize : mean + fused dual GEMM (agg@Wl + bl + h@Wr) via
//              V_WMMA_F32_16X16X4_F32 (full f32 precision matrix pipe)
// ---------------------------------------------------------------------------

static constexpr int NU = 100000;   // multiple of 16 (stripe-exact)
static constexpr int NT = 200000;   // multiple of 16
static constexpr int NG = 20000;    // multiple of 16
static constexpr int XD = 32;
static constexpr int HD = 64;
static constexpr int E_UT = 2000000;
static constexpr int E_TG = 500000;

typedef __attribute__((ext_vector_type(2))) float v2f;
typedef __attribute__((ext_vector_type(8))) float v8f;

// ---------------------------------------------------------------------------
// encode: h[n][0:32] = emb[node_id[n]][:]; h[n][32:64] = relu(x[n] @ W + b)
// One wave per node; lane = output column. x row value shared via shfl.
// ---------------------------------------------------------------------------
__global__ void encode_kernel(const float* __restrict__ x,    // [N, 32]
                              const float* __restrict__ emb,  // [N, 32]
                              const int*   __restrict__ nid,  // [N]
                              const float* __restrict__ W,    // [32, 32] row-major (in,out)
                              const float* __restrict__ b,    // [32]
                              float* __restrict__ h,          // [N, 64]
                              int N) {
    __shared__ float sW[XD * XD];
    __shared__ float sb[XD];
    int tid = threadIdx.x;                       // 256 threads = 8 waves
    for (int i = tid; i < XD * XD; i += 256) sW[i] = W[i];
    if (tid < XD) sb[tid] = b[tid];
    __syncthreads();

    int node = blockIdx.x * 8 + (tid >> 5);      // one wave per node
    int col  = tid & 31;
    if (node >= N) return;

    float xv = x[(size_t)node * XD + col];       // lane's element of the x row
    float acc = 0.0f;
#pragma unroll
    for (int k = 0; k < XD; ++k) {
        float xk = __shfl(xv, k, 32);            // broadcast x[node][k]
        acc = fmaf(xk, sW[k * XD + col], acc);
    }
    acc = fmaxf(acc + sb[col], 0.0f);

    int id = nid[node];
    float ev = emb[(size_t)id * XD + col];
    float* hrow = h + (size_t)node * HD;
    hrow[col]      = ev;                          // embedding half
    hrow[XD + col] = acc;                         // relu(feature) half
}

// ---------------------------------------------------------------------------
// scatter: for each edge e: agg[dst[e]] += h_src[src[e]]; cnt[dst[e]] += 1
// One wave per edge: 32 lanes x float2 = one coalesced 256B row gather,
// then hardware f32 atomics (global_atomic_add_f32, no return -> STOREcnt).
// Node tables (~82 MB total) are L2-resident on the 192 MB L2.
// ---------------------------------------------------------------------------
__global__ void scatter_kernel(const float* __restrict__ hsrc, // [Nsrc, 64]
                               const int*   __restrict__ src,
                               const int*   __restrict__ dst,
                               float* __restrict__ agg,        // [Ndst, 64]
                               float* __restrict__ cnt,        // [Ndst]
                               int E) {
    int e = blockIdx.x * 8 + (threadIdx.x >> 5);
    if (e >= E) return;
    int lane = threadIdx.x & 31;

    int s = src[e];                               // wave-uniform -> scalarized
    int d = dst[e];

    const float2* srow = (const float2*)(hsrc + (size_t)s * HD);
    float2 v = srow[lane];

    float* arow = agg + (size_t)d * HD + lane * 2;
    unsafeAtomicAdd(arow,     v.x);
    unsafeAtomicAdd(arow + 1, v.y);
    if (lane == 0) unsafeAtomicAdd(cnt + d, 1.0f);
}

// ---------------------------------------------------------------------------
// finalize: out[r] = relu?( scale * ( (agg[r]/max(cnt,1)) @ Wl + bl
//                                     + hroot[r] @ Wr )  [+ out[r] if accum] )
//
// One wave computes a 16x64 output stripe with V_WMMA_F32_16X16X4_F32:
//   K=64 as 16 steps of K=4; per step 2 WMMAs (Wl path, Wr path) into the
//   same 4 N-tile accumulators (4 x v8f = 32 VGPRs).
//
// Fragment layouts (ISA 7.12.2, f32):
//   A 16x4 : lane L -> row L%16, K pair 4*kc + 2*(L/16)   (v2f, global b64)
//   B 4x16 : lane L -> col L%16, K pair       2*(L/16)    (v2f, LDS b64)
//   C/D    : VGPR j -> row j + 8*(L/16), col L%16
//
// Weights are staged in LDS pre-swizzled as K-pairs so each B fragment is a
// single aligned ds_load_b64 (no v_mov repacking):
//   sW[kp*PSTR + 2*n + klo] = W[2*kp + klo][n],  kp = 0..31
// PSTR = 160 floats => kp-row stride is 32 banks (mod 64), so the two
// half-wave chunks (kp, kp+1) hit disjoint 32-bank sets (no 2-way conflict).
//
// Requires N % 16 == 0 (true for NU/NT/NG) -> no per-row predication.
// ---------------------------------------------------------------------------
__global__ void finalize_kernel(const float* __restrict__ agg,   // [N, 64]
                                const float* __restrict__ cnt,   // [N]
                                const float* __restrict__ hroot, // [N, 64]
                                const float* __restrict__ Wl,    // [64, 64]
                                const float* __restrict__ bl,    // [64]
                                const float* __restrict__ Wr,    // [64, 64]
                                float* __restrict__ out,         // [N, 64]
                                int N, float scale, int accum, int do_relu) {
    constexpr int PSTR = 160;                     // 128 data + 32 pad floats
    __shared__ __align__(16) float sWl[32 * PSTR];
    __shared__ __align__(16) float sWr[32 * PSTR];
    __shared__ float sb[HD];

    int tid = threadIdx.x;                        // 256 threads = 8 waves
    for (int i = tid; i < HD * HD; i += 256) {    // swizzle into K-pair layout
        int k  = i >> 6, n = i & 63;
        int kp = k >> 1, klo = k & 1;
        sWl[kp * PSTR + 2 * n + klo] = Wl[i];
        sWr[kp * PSTR + 2 * n + klo] = Wr[i];
    }
    if (tid < HD) sb[tid] = bl[tid];
    __syncthreads();

    int wave = tid >> 5;
    int lane = tid & 31;
    int row0 = (blockIdx.x * 8 + wave) * 16;      // this wave's 16-row stripe
    if (row0 >= N) return;                        // wave-uniform exit

    int mrow  = lane & 15;                        // A row / B,C col within tile
    int khalf = lane >> 4;                        // K-pair selector
    int grow  = row0 + mrow;                      // always < N (N % 16 == 0)

    float rinv = 1.0f / fmaxf(cnt[grow], 1.0f);   // mean divisor, folded into A
    const v2f* arow = (const v2f*)(agg   + (size_t)grow * HD);
    const v2f* hrow = (const v2f*)(hroot + (size_t)grow * HD);

    // Bulk-load all A fragments up front (32 x global_load_b64, 64 VGPRs);
    // mean scaling folds into packed multiplies.
    v2f aA[16], aR[16];
#pragma unroll
    for (int kc = 0; kc < 16; ++kc) {
        aA[kc] = arow[kc * 2 + khalf] * rinv;
        aR[kc] = hrow[kc * 2 + khalf];
    }

    v8f c[4] = {v8f{}, v8f{}, v8f{}, v8f{}};

#pragma unroll
    for (int kc = 0; kc < 16; ++kc) {
        int kp = kc * 2 + khalf;
        const v2f* bLrow = (const v2f*)(sWl + kp * PSTR);
        const v2f* bRrow = (const v2f*)(sWr + kp * PSTR);
#pragma unroll
        for (int nt = 0; nt < 4; ++nt) {
            int ncol = nt * 16 + mrow;
            v2f bL = bLrow[ncol];                 // single ds_load_b64
            v2f bR = bRrow[ncol];
            c[nt] = __builtin_amdgcn_wmma_f32_16x16x4_f32(
                        false, aA[kc], false, bL, (short)0, c[nt], false, false);
            c[nt] = __builtin_amdgcn_wmma_f32_16x16x4_f32(
                        false, aR[kc], false, bR, (short)0, c[nt], false, false);
        }
    }

    // Epilogue: bias, relation-mean scale, optional accumulate + ReLU.
#pragma unroll
    for (int nt = 0; nt < 4; ++nt) {
        int ncol = nt * 16 + mrow;
        float bv = sb[ncol];
#pragma unroll
        for (int j = 0; j < 8; ++j) {
            int orow = row0 + j + 8 * khalf;
            float val = (c[nt][j] + bv) * scale;
            float* op = out + (size_t)orow * HD + ncol;
            if (accum)   val += *op;
            if (do_relu) val  = fmaxf(val, 0.0f);
            *op = val;
        }
    }
}

// ---------------------------------------------------------------------------
// Host-side orchestration.
//
// Input flattening (top-level in setup_inputs() order, nested pytrees
// flattened JAX-style with sorted dict keys):
//   0-2   node_id_{user,track,tag}        3-5   x_{user,track,tag}
//   6-9   src/dst u2t, src/dst t2u        10-13 src/dst t2g, src/dst g2t
//   14-16 emb_{user,track,tag}
//   17-22 feat_params sorted(tag,track,user): (W,b) each
//   23-34 sage layer0 sorted(g2t,t2g,t2u,u2t): (Wl,bl,Wr) each
//   35-46 sage layer1 same order
// Output: u [NU*64] | t [NT*64] | g [NG*64] (f32).
// ---------------------------------------------------------------------------
extern "C" void kernel_launch(void* const* d_in, const int* in_sizes, int n_in,
                              void* d_out, int out_size, void* d_ws, size_t ws_size,
                              hipStream_t stream) {
    const int*   nid_u = (const int*)d_in[0];
    const int*   nid_t = (const int*)d_in[1];
    const int*   nid_g = (const int*)d_in[2];
    const float* x_u   = (const float*)d_in[3];
    const float* x_t   = (const float*)d_in[4];
    const float* x_g   = (const float*)d_in[5];
    const int*   src_u2t = (const int*)d_in[6];
    const int*   dst_u2t = (const int*)d_in[7];
    const int*   src_t2u = (const int*)d_in[8];
    const int*   dst_t2u = (const int*)d_in[9];
    const int*   src_t2g = (const int*)d_in[10];
    const int*   dst_t2g = (const int*)d_in[11];
    const int*   src_g2t = (const int*)d_in[12];
    const int*   dst_g2t = (const int*)d_in[13];
    const float* emb_u = (const float*)d_in[14];
    const float* emb_t = (const float*)d_in[15];
    const float* emb_g = (const float*)d_in[16];
    // feat_params (sorted keys: tag, track, user)
    const float* Wf_g = (const float*)d_in[17];
    const float* bf_g = (const float*)d_in[18];
    const float* Wf_t = (const float*)d_in[19];
    const float* bf_t = (const float*)d_in[20];
    const float* Wf_u = (const float*)d_in[21];
    const float* bf_u = (const float*)d_in[22];

    // sage params: per layer, sorted rel keys: g2t(0), t2g(1), t2u(2), u2t(3)
    auto sageWl = [&](int layer, int rel) { return (const float*)d_in[23 + layer * 12 + rel * 3 + 0]; };
    auto sageBl = [&](int layer, int rel) { return (const float*)d_in[23 + layer * 12 + rel * 3 + 1]; };
    auto sageWr = [&](int layer, int rel) { return (const float*)d_in[23 + layer * 12 + rel * 3 + 2]; };
    const int R_G2T = 0, R_T2G = 1, R_T2U = 2, R_U2T = 3;

    // Workspace layout (floats): hA | hB | agg(NT*64) | cnt(NT)
    const size_t H_TOTAL = (size_t)(NU + NT + NG) * HD;
    float* hA  = (float*)d_ws;
    float* hB  = hA + H_TOTAL;
    float* agg = hB + H_TOTAL;
    float* cnt = agg + (size_t)NT * HD;

    const size_t OU = 0;
    const size_t OT = (size_t)NU * HD;
    const size_t OG = (size_t)(NU + NT) * HD;

    // ---- encode all three node types into hA -----------------------------
    encode_kernel<<<(NU + 7) / 8, 256, 0, stream>>>(x_u, emb_u, nid_u, Wf_u, bf_u, hA + OU, NU);
    encode_kernel<<<(NT + 7) / 8, 256, 0, stream>>>(x_t, emb_t, nid_t, Wf_t, bf_t, hA + OT, NT);
    encode_kernel<<<(NG + 7) / 8, 256, 0, stream>>>(x_g, emb_g, nid_g, Wf_g, bf_g, hA + OG, NG);

    float* outbuf = (float*)d_out;

    for (int layer = 0; layer < 2; ++layer) {
        const float* hin  = (layer == 0) ? hA : hB;
        float*       hout = (layer == 0) ? hB : outbuf;
        int relu = (layer == 0) ? 1 : 0;

        // ---- t2u : track -> user ----------------------------------------
        hipMemsetAsync(agg, 0, (size_t)NU * HD * sizeof(float), stream);
        hipMemsetAsync(cnt, 0, (size_t)NU * sizeof(float), stream);
        scatter_kernel<<<(E_UT + 7) / 8, 256, 0, stream>>>(hin + OT, src_t2u, dst_t2u, agg, cnt, E_UT);
        finalize_kernel<<<(NU + 127) / 128, 256, 0, stream>>>(
            agg, cnt, hin + OU, sageWl(layer, R_T2U), sageBl(layer, R_T2U), sageWr(layer, R_T2U),
            hout + OU, NU, 1.0f, /*accum=*/0, relu);

        // ---- u2t : user -> track (first half of relation mean) ----------
        hipMemsetAsync(agg, 0, (size_t)NT * HD * sizeof(float), stream);
        hipMemsetAsync(cnt, 0, (size_t)NT * sizeof(float), stream);
        scatter_kernel<<<(E_UT + 7) / 8, 256, 0, stream>>>(hin + OU, src_u2t, dst_u2t, agg, cnt, E_UT);
        finalize_kernel<<<(NT + 127) / 128, 256, 0, stream>>>(
            agg, cnt, hin + OT, sageWl(layer, R_U2T), sageBl(layer, R_U2T), sageWr(layer, R_U2T),
            hout + OT, NT, 0.5f, /*accum=*/0, /*relu=*/0);

        // ---- g2t : tag -> track (second half; accumulate, then relu) ----
        hipMemsetAsync(agg, 0, (size_t)NT * HD * sizeof(float), stream);
        hipMemsetAsync(cnt, 0, (size_t)NT * sizeof(float), stream);
        scatter_kernel<<<(E_TG + 7) / 8, 256, 0, stream>>>(hin + OG, src_g2t, dst_g2t, agg, cnt, E_TG);
        finalize_kernel<<<(NT + 127) / 128, 256, 0, stream>>>(
            agg, cnt, hin + OT, sageWl(layer, R_G2T), sageBl(layer, R_G2T), sageWr(layer, R_G2T),
            hout + OT, NT, 0.5f, /*accum=*/1, relu);

        // ---- t2g : track -> tag -----------------------------------------
        hipMemsetAsync(agg, 0, (size_t)NG * HD * sizeof(float), stream);
        hipMemsetAsync(cnt, 0, (size_t)NG * sizeof(float), stream);
        scatter_kernel<<<(E_TG + 7) / 8, 256, 0, stream>>>(hin + OT, src_t2g, dst_t2g, agg, cnt, E_TG);
        finalize_kernel<<<(NG + 127) / 128, 256, 0, stream>>>(
            agg, cnt, hin + OG, sageWl(layer, R_T2G), sageBl(layer, R_T2G), sageWr(layer, R_T2G),
            hout + OG, NG, 1.0f, /*accum=*/0, relu);
    }
}